// Group_50044958933387
// MI455X (gfx1250) — compile-verified
//
#include <hip/hip_runtime.h>
#include <cstdint>

// Problem constants (from reference)
#define B_      16
#define N_      8192
#define C_      6
#define G_      512      // NUM_GROUP
#define S_      32       // GROUP_SIZE
#define BIGF    1.0e10f

// Output layout: [neighborhood (B*G*S*C)] [center (B*G*C)] [nn_idx (B*G*S)]
#define NEI_ELEMS  (B_*G_*S_*C_)     // 1,572,864
#define CEN_OFF    NEI_ELEMS
#define CEN_ELEMS  (B_*G_*C_)        //    49,152
#define IDX_OFF    (CEN_OFF + CEN_ELEMS)
#define IDX_ELEMS  (B_*G_*S_)        //   262,144

typedef float v2f __attribute__((ext_vector_type(2)));
typedef float v8f __attribute__((ext_vector_type(8)));

// ---------------------------------------------------------------------------
// Kernel 1: farthest point sampling, one workgroup per batch.
// 1024 threads = 32 wave32s. Points packed (x,y,z,running_min_dist) as one
// float4 in LDS (128KB of the 320KB WGP LDS) -> one ds_load_b128 + one
// ds_store_b32 per point per sweep. 512 sequential argmax sweeps; per-wave
// shuffle argmax then wave0 shuffle reduce across the 32 waves.
// ---------------------------------------------------------------------------
__global__ __launch_bounds__(1024) void fps_kernel(const float* __restrict__ pc,
                                                   float* __restrict__ center_out) {
    __shared__ float4 pts[N_];              // x, y, z, dist
    __shared__ float  rv[32];
    __shared__ int    ri[32];
    __shared__ int    s_last;

    const int b   = blockIdx.x;
    const int tid = threadIdx.x;
    const float* base = pc + (size_t)b * N_ * C_;
    float* cdst = center_out + (size_t)b * G_ * C_;

    for (int n = tid; n < N_; n += 1024) {
        const float* p = base + (size_t)n * C_;
        pts[n] = make_float4(p[0], p[1], p[2], BIGF);
    }
    if (tid == 0) {
        s_last = 0;
        // rep_idx[0] == 0 per reference (scan emits pre-update 'last')
        for (int c = 0; c < C_; ++c) cdst[c] = base[c];
    }
    __syncthreads();

    int last = 0;
    for (int k = 1; k < G_; ++k) {
        const float4 lp = pts[last];        // broadcast read
        float bestV = -1.0f;
        int   bestI = 0x7fffffff;
        for (int n = tid; n < N_; n += 1024) {
            float4 q = pts[n];
            const float dx = q.x - lp.x, dy = q.y - lp.y, dz = q.z - lp.z;
            const float d  = dx*dx + dy*dy + dz*dz;
            float dd = fminf(q.w, d);
            pts[n].w = dd;
            if (dd > bestV) { bestV = dd; bestI = n; }   // strided n increasing -> first-max kept
        }
        // wave32 argmax reduce (tie -> smaller index, matching jnp.argmax)
        #pragma unroll
        for (int off = 16; off > 0; off >>= 1) {
            const float ov = __shfl_xor(bestV, off, 32);
            const int   oi = __shfl_xor(bestI, off, 32);
            if (ov > bestV || (ov == bestV && oi < bestI)) { bestV = ov; bestI = oi; }
        }
        if ((tid & 31) == 0) { rv[tid >> 5] = bestV; ri[tid >> 5] = bestI; }
        __syncthreads();
        if (tid < 32) {                     // wave0: parallel cross-wave argmax
            float bv = rv[tid]; int bi = ri[tid];
            #pragma unroll
            for (int off = 16; off > 0; off >>= 1) {
                const float ov = __shfl_xor(bv, off, 32);
                const int   oi = __shfl_xor(bi, off, 32);
                if (ov > bv || (ov == bv && oi < bi)) { bv = ov; bi = oi; }
            }
            if (tid == 0) {
                s_last = bi;
                const float* p = base + (size_t)bi * C_;
                float* cd = cdst + (size_t)k * C_;
                cd[0]=p[0]; cd[1]=p[1]; cd[2]=p[2]; cd[3]=p[3]; cd[4]=p[4]; cd[5]=p[5];
            }
        }
        __syncthreads();
        last = s_last;
    }
}

// ---------------------------------------------------------------------------
// LDS-resident top-32 insertion (ascending dist, stable w.r.t. index like
// lax.top_k on -d2). Rarely taken: ~32*ln(N/32) ~ 180 inserts per row.
// ---------------------------------------------------------------------------
__device__ __forceinline__ void top32_insert(float* __restrict__ tv,
                                             int*   __restrict__ ti,
                                             float v, int idx) {
    int pos = S_ - 1;
    while (pos > 0 && tv[pos - 1] > v) {
        tv[pos] = tv[pos - 1];
        ti[pos] = ti[pos - 1];
        --pos;
    }
    tv[pos] = v;
    ti[pos] = idx;
}

// ---------------------------------------------------------------------------
// Kernel 2: kNN via V_WMMA_F32_16X16X4_F32.
// score = |t|^2 - 2 q.t  (the |q|^2 term is row-constant, irrelevant to top-k)
//   A (16x4): rows = centers, [-2qx, -2qy, -2qz, 1]
//   B (4x16): cols = points,  [ tx,  ty,  tz, |t|^2]
// One block = 16 centers of one batch; 128 threads = 4 waves.
// Per 512-point chunk each wave batch-loads its 8 point-tiles into registers
// (8 back-to-back global_load_b96, waits amortized), then runs 8 WMMAs with
// LDS writeback. Selection: 16 row owners, float4 scan with min-of-4
// early-out; top-32 lists live in LDS (bank-spread stride 33).
// ---------------------------------------------------------------------------
#define KNN_CHUNK 512
#define KNN_TPW   8                       // tiles per wave per chunk: 512/16/4
__global__ __launch_bounds__(128) void knn_kernel(const float* __restrict__ pc,
                                                  const float* __restrict__ centers,
                                                  int*   __restrict__ nn_out,
                                                  float* __restrict__ nnf_out) {
    __shared__ float sc[16][KNN_CHUNK + 4];   // stride 516 floats: 16B-aligned rows, banks spread
    __shared__ float tV[16][33];              // per-owner top-32 values (+pad -> distinct banks)
    __shared__ int   tI[16][33];              // per-owner top-32 indices

    const int gtile = blockIdx.x;           // 0..31
    const int b     = blockIdx.y;           // 0..15
    const int tid   = threadIdx.x;          // 0..127
    const int wave  = tid >> 5;
    const int lane  = tid & 31;
    const int col   = lane & 15;
    const int hi    = lane >> 4;            // 0: K0/K1 half, 1: K2/K3 half

    // A fragment (constant for the whole block): 16x4 f32, ISA layout
    const int m = gtile * 16 + col;
    const float* cp = centers + (size_t)(b * G_ + m) * C_;
    const float cx = cp[0], cy = cp[1], cz = cp[2];
    v2f afrag;
    afrag.x = hi ? (-2.0f * cz) : (-2.0f * cx);
    afrag.y = hi ? 1.0f         : (-2.0f * cy);

    if (tid < 16) {                         // each owner inits only its own row: no barrier needed
        for (int i = 0; i < S_; ++i) { tV[tid][i] = BIGF; tI[tid][i] = 0; }
    }

    const float* pbase = pc + (size_t)b * N_ * C_;

    for (int chunk = 0; chunk < N_; chunk += KNN_CHUNK) {
        // --- batch-load 8 point tiles (24 VGPRs), then 8 WMMAs; EXEC all-1s ---
        float xa[KNN_TPW], ya[KNN_TPW], za[KNN_TPW];
        #pragma unroll
        for (int i = 0; i < KNN_TPW; ++i) {
            const int t = wave + i * 4;
            const float* pp = pbase + (size_t)(chunk + t * 16 + col) * C_;
            xa[i] = pp[0]; ya[i] = pp[1]; za[i] = pp[2];
        }
        #pragma unroll
        for (int i = 0; i < KNN_TPW; ++i) {
            const int t = wave + i * 4;
            v2f bfrag;
            bfrag.x = hi ? za[i] : xa[i];
            bfrag.y = hi ? (xa[i]*xa[i] + ya[i]*ya[i] + za[i]*za[i]) : ya[i];
            v8f acc = {};
            v8f d = __builtin_amdgcn_wmma_f32_16x16x4_f32(
                        false, afrag, false, bfrag, (short)0, acc, false, false);
            const int rbase = hi * 8;       // C/D layout: lanes 0-15 -> M=r, lanes 16-31 -> M=8+r
            const int cc    = t * 16 + col;
            #pragma unroll
            for (int r = 0; r < 8; ++r) sc[rbase + r][cc] = d[r];
        }
        __syncthreads();

        // --- selection: 16 row owners, float4 scan + threshold early-out ---
        if (tid < 16) {
            float* tv = tV[tid];
            int*   ti = tI[tid];
            const float4* row = (const float4*)(&sc[tid][0]);
            for (int j4 = 0; j4 < KNN_CHUNK / 4; ++j4) {
                const float4 v = row[j4];
                const float mn = fminf(fminf(v.x, v.y), fminf(v.z, v.w));
                float worst = tv[S_ - 1];
                if (mn < worst) {
                    const int jb = chunk + j4 * 4;
                    if (v.x < worst) { top32_insert(tv, ti, v.x, jb + 0); worst = tv[S_ - 1]; }
                    if (v.y < worst) { top32_insert(tv, ti, v.y, jb + 1); worst = tv[S_ - 1]; }
                    if (v.z < worst) { top32_insert(tv, ti, v.z, jb + 2); worst = tv[S_ - 1]; }
                    if (v.w < worst) { top32_insert(tv, ti, v.w, jb + 3); }
                }
            }
        }
        __syncthreads();
    }

    if (tid < 16) {
        const size_t o = ((size_t)(b * G_ + m)) * S_;
        for (int k = 0; k < S_; ++k) {
            const int iv = tI[tid][k];
            nn_out[o + k]  = iv;
            nnf_out[o + k] = (float)iv;
        }
    }
}

// ---------------------------------------------------------------------------
// Kernel 3: gather neighborhood, subtract center xyz, keep channels 3..5.
// One thread per (b, g, s) element. Pure streaming; L2-resident data.
// ---------------------------------------------------------------------------
__global__ __launch_bounds__(256) void gather_kernel(const float* __restrict__ pc,
                                                     const float* __restrict__ centers,
                                                     const int* __restrict__ nn_idx,
                                                     float* __restrict__ nei_out) {
    const int t = blockIdx.x * blockDim.x + threadIdx.x;
    if (t >= IDX_ELEMS) return;
    const int b = t / (G_ * S_);
    const int g = (t / S_) % G_;
    const int idx = nn_idx[t];

    const float* p = pc + ((size_t)b * N_ + idx) * C_;
    const float* c = centers + (size_t)(b * G_ + g) * C_;
    float* o = nei_out + (size_t)t * C_;
    o[0] = p[0] - c[0];
    o[1] = p[1] - c[1];
    o[2] = p[2] - c[2];
    o[3] = p[3];
    o[4] = p[4];
    o[5] = p[5];
}

// ---------------------------------------------------------------------------
extern "C" void kernel_launch(void* const* d_in, const int* in_sizes, int n_in,
                              void* d_out, int out_size, void* d_ws, size_t ws_size,
                              hipStream_t stream) {
    const float* pc = (const float*)d_in[0];
    float* out     = (float*)d_out;
    float* center  = out + CEN_OFF;
    float* idx_f   = out + IDX_OFF;
    int*   nn_ws   = (int*)d_ws;          // 1 MB of scratch for integer indices

    fps_kernel<<<B_, 1024, 0, stream>>>(pc, center);
    knn_kernel<<<dim3(G_ / 16, B_), 128, 0, stream>>>(pc, center, nn_ws, idx_f);
    gather_kernel<<<(IDX_ELEMS + 255) / 256, 256, 0, stream>>>(pc, center, nn_ws, out);
}